// GraphSAGELayer_55490977464425
// MI455X (gfx1250) — compile-verified
//
#include <hip/hip_runtime.h>

typedef float v2f __attribute__((ext_vector_type(2)));
typedef float v8f __attribute__((ext_vector_type(8)));

#define IN_DIM  128
#define OUT_DIM 128

// ---------------------------------------------------------------------------
// Kernel 1: zero the workspace (deg[N] + agg[N*128]) via grid-stride scalar float stores.
// ---------------------------------------------------------------------------
__global__ void sage_init_ws(float* __restrict__ ws, long long n_floats) {
    long long i = (long long)blockIdx.x * blockDim.x + threadIdx.x;
    long long stride = (long long)gridDim.x * blockDim.x;
    for (; i < n_floats; i += stride) ws[i] = 0.0f;
}

// ---------------------------------------------------------------------------
// Kernel 2: bidirectional scatter-add. One wave32 per undirected edge.
//   agg[dst] += x[src];  agg[src] += x[dst];  deg[src]++; deg[dst]++
// 128 floats/row -> 4 coalesced lane-sweeps of global_atomic_add_f32.
// agg + x both live in the 192MB L2, so these atomics resolve on-chip.
// ---------------------------------------------------------------------------
__global__ void __launch_bounds__(256)
sage_scatter(const float* __restrict__ x,
             const long long* __restrict__ ei,   // [2, E] row-major int64
             float* __restrict__ agg,
             float* __restrict__ deg,
             int n_edges) {
    int wid  = (blockIdx.x * blockDim.x + threadIdx.x) >> 5;  // wave id = edge id
    int lane = threadIdx.x & 31;
    if (wid >= n_edges) return;

    long long s = ei[wid];
    long long d = ei[n_edges + wid];

    if (lane == 0)      atomicAdd(&deg[s], 1.0f);
    else if (lane == 1) atomicAdd(&deg[d], 1.0f);

    const float* xs = x + s * IN_DIM;
    const float* xd = x + d * IN_DIM;
    float* as = agg + s * IN_DIM;
    float* ad = agg + d * IN_DIM;

#pragma unroll
    for (int j = 0; j < 4; ++j) {
        int c = lane + 32 * j;           // coalesced across the wave
        atomicAdd(&ad[c], xs[c]);        // forward edge
        atomicAdd(&as[c], xd[c]);        // reverse edge
    }
}

// ---------------------------------------------------------------------------
// Kernel 3: out = relu(x @ Wself^T + (agg/deg) @ Wneigh^T + bias)
// Full-fp32 WMMA (V_WMMA_F32_16X16X4_F32): wave computes a 16x16 tile,
// block (8 waves) computes 16 rows x 128 cols. K=128 -> 32+32 WMMAs into a
// single v8f accumulator; degree normalization folded into the A-load.
// ---------------------------------------------------------------------------
__global__ void __launch_bounds__(256)
sage_gemm(const float* __restrict__ x,
          const float* __restrict__ agg,
          const float* __restrict__ deg,
          const float* __restrict__ Wself,    // [128,128] row-major
          const float* __restrict__ Wneigh,   // [128,128] row-major
          const float* __restrict__ bias,     // [128]
          float* __restrict__ out) {
    const int wave = threadIdx.x >> 5;   // 0..7 -> 16-col tile
    const int lane = threadIdx.x & 31;
    const int row0 = blockIdx.x * 16;
    const int col0 = wave * 16;
    const int m  = lane & 15;            // A row (M) / B row (N) for this lane
    const int kh = lane >> 4;            // K half-select: k-offset 2*kh

    const float* xrow = x      + (long long)(row0 + m) * IN_DIM;
    const float* arow = agg    + (long long)(row0 + m) * IN_DIM;
    const float* wsr  = Wself  + (long long)(col0 + m) * IN_DIM;  // B = W^T
    const float* wnr  = Wneigh + (long long)(col0 + m) * IN_DIM;

    const float rdeg = 1.0f / fmaxf(deg[row0 + m], 1.0f);

    v8f c = {0.f, 0.f, 0.f, 0.f, 0.f, 0.f, 0.f, 0.f};

    // x @ Wself^T : 32 x V_WMMA_F32_16X16X4_F32
#pragma unroll 8
    for (int k = 0; k < IN_DIM; k += 4) {
        v2f a = *(const v2f*)(xrow + k + 2 * kh);   // A[m, k+2kh .. +1]
        v2f b = *(const v2f*)(wsr  + k + 2 * kh);   // B[k+2kh .. +1, m]
        c = __builtin_amdgcn_wmma_f32_16x16x4_f32(
                false, a, false, b, (short)0, c, false, false);
    }
    // (agg/deg) @ Wneigh^T : 32 more, same accumulator
#pragma unroll 8
    for (int k = 0; k < IN_DIM; k += 4) {
        v2f a = *(const v2f*)(arow + k + 2 * kh);
        a.x *= rdeg;
        a.y *= rdeg;
        v2f b = *(const v2f*)(wnr + k + 2 * kh);
        c = __builtin_amdgcn_wmma_f32_16x16x4_f32(
                false, a, false, b, (short)0, c, false, false);
    }

    // Epilogue: bias + relu, write 16x16 tile.
    // C/D layout: VGPR r -> row = row0 + r + 8*(lane/16), col = col0 + lane%16
    const int   nCol  = col0 + m;
    const float bcol  = bias[nCol];
    const int   mBase = row0 + kh * 8;
#pragma unroll
    for (int r = 0; r < 8; ++r) {
        float v = fmaxf(c[r] + bcol, 0.0f);
        out[(long long)(mBase + r) * OUT_DIM + nCol] = v;
    }
}

// ---------------------------------------------------------------------------
extern "C" void kernel_launch(void* const* d_in, const int* in_sizes, int n_in,
                              void* d_out, int out_size, void* d_ws, size_t ws_size,
                              hipStream_t stream) {
    const float*     x      = (const float*)d_in[0];
    const long long* ei     = (const long long*)d_in[1];   // int64 per reference
    // d_in[2] = num_nodes scalar (value derived from in_sizes instead)
    const float*     Wself  = (const float*)d_in[3];
    const float*     Wneigh = (const float*)d_in[4];
    const float*     bias   = (const float*)d_in[5];
    float*           out    = (float*)d_out;

    const int N = in_sizes[0] / IN_DIM;   // 100000
    const int E = in_sizes[1] / 2;        // 800000

    float* deg = (float*)d_ws;            // [N]
    float* agg = deg + N;                 // [N * 128]

    // 1) zero deg + agg
    const long long n_ws_floats = (long long)N * (IN_DIM + 1);
    sage_init_ws<<<2048, 256, 0, stream>>>((float*)d_ws, n_ws_floats);

    // 2) bidirectional scatter: one wave per undirected edge (8 waves/block)
    const int scatter_blocks = (E + 7) / 8;
    sage_scatter<<<scatter_blocks, 256, 0, stream>>>(x, ei, agg, deg, E);

    // 3) fused dual-GEMM + bias + relu (fp32 WMMA)
    const int gemm_blocks = (N + 15) / 16; // 6250
    sage_gemm<<<gemm_blocks, 256, 0, stream>>>(x, agg, deg, Wself, Wneigh, bias, out);
}